// Neuron_75222057222208
// MI455X (gfx1250) — compile-verified
//
#include <hip/hip_runtime.h>

// LIF eval-mode scan: x[B,T,N] -> o[B,T,N], sequential over T per (b,n).
//   u = 0.5*u + x_t ; o = (u > 1.0) ; u = o ? 0 : u
// Memory-bound (268 MB once-through @ 23.3 TB/s). One thread per float4
// column, 16-deep load batching for MLP, NT hints, global_prefetch of the
// next batch.

typedef float v4f __attribute__((ext_vector_type(4)));

#define T_STEPS   128
#define N_NEUR    8192
#define TAU_F     0.5f
#define THRESH_F  1.0f
#define BATCH     16

__global__ __launch_bounds__(256) void lif_scan_kernel(
    const float* __restrict__ x, float* __restrict__ o, int ncols) {
  const int tid = blockIdx.x * blockDim.x + threadIdx.x;
  if (tid >= ncols) return;

  const int colsPerB = N_NEUR / 4;              // 2048 v4-columns per batch row
  const int b = tid / colsPerB;
  const int c = tid - b * colsPerB;

  const size_t base = (size_t)b * T_STEPS * N_NEUR + (size_t)c * 4;
  const v4f* __restrict__ xp = (const v4f*)(x + base);
  v4f* __restrict__ op = (v4f*)(o + base);
  const size_t strideV4 = N_NEUR / 4;           // one timestep, in v4f units

  v4f u = {0.0f, 0.0f, 0.0f, 0.0f};

  for (int t0 = 0; t0 < T_STEPS; t0 += BATCH) {
    // Batch the loads: 16 outstanding global_load_b128 (NT) per wave.
    v4f xin[BATCH];
#pragma unroll
    for (int j = 0; j < BATCH; ++j)
      xin[j] = __builtin_nontemporal_load(xp + (size_t)(t0 + j) * strideV4);

    // Prefetch next batch's cachelines while the recurrence runs.
    if (t0 + BATCH < T_STEPS) {
#pragma unroll
      for (int j = 0; j < BATCH; ++j)
        __builtin_prefetch((const void*)(xp + (size_t)(t0 + BATCH + j) * strideV4), 0, 1);
    }

    // Sequential LIF recurrence over the batched timesteps.
#pragma unroll
    for (int j = 0; j < BATCH; ++j) {
      v4f spk;
#pragma unroll
      for (int k = 0; k < 4; ++k) {
        float uu = __builtin_fmaf(TAU_F, u[k], xin[j][k]);  // leaky integrate
        bool fire = (uu > THRESH_F);                        // spike(u - 1) == (u > 1)
        spk[k] = fire ? 1.0f : 0.0f;
        u[k]   = fire ? 0.0f : uu;                          // hard reset
      }
      __builtin_nontemporal_store(spk, op + (size_t)(t0 + j) * strideV4);
    }
  }
}

extern "C" void kernel_launch(void* const* d_in, const int* in_sizes, int n_in,
                              void* d_out, int out_size, void* d_ws, size_t ws_size,
                              hipStream_t stream) {
  const float* x = (const float*)d_in[0];
  float* o = (float*)d_out;

  // total = B*T*N ; v4-columns = B*N/4 = total / (T*4)
  const int total = in_sizes[0];
  const int ncols = total / (T_STEPS * 4);

  const int block = 256;
  const int grid = (ncols + block - 1) / block;
  lif_scan_kernel<<<grid, block, 0, stream>>>(x, o, ncols);
}